// Erosion_6339371728979
// MI455X (gfx1250) — compile-verified
//
#include <hip/hip_runtime.h>

// ---------------------------------------------------------------------------
// 5x5 erosion (sliding-window min) over NCHW f32, geodesic pad = 1e4.
// HBM-bound: 256 MB traffic => ~11us floor at 23.3 TB/s.
// CDNA5 path: GLOBAL_LOAD_ASYNC_TO_LDS_B128 staging (ASYNCcnt) +
// s_wait_asynccnt + float4 LDS separable min + b128 non-temporal stores.
// ---------------------------------------------------------------------------

#define HW_  1024
#define P    2
#define TW   128             // output tile width
#define TH   16              // output tile height
#define LW   136             // staged columns: [tx0-4, tx0+132) -> 4-aligned chunks
#define LH   (TH + 2 * P)    // 20 staged rows
#define NCHUNK (LW / 4)      // 34 b128 chunks per row
#define MAXV 1.0e4f
#define NTHREADS 256

#if defined(__HIP_DEVICE_COMPILE__) && defined(__gfx1250__) && \
    __has_builtin(__builtin_amdgcn_global_load_async_to_lds_b128)
#define ASYNC_MODE 2   // b128 DMA
#elif defined(__HIP_DEVICE_COMPILE__) && defined(__gfx1250__) && \
    __has_builtin(__builtin_amdgcn_global_load_async_to_lds_b32)
#define ASYNC_MODE 1   // b32 DMA
#else
#define ASYNC_MODE 0   // plain loads
#endif

// native clang vector types (usable with amdgcn builtins)
typedef float f4 __attribute__((ext_vector_type(4)));
typedef int   i4 __attribute__((ext_vector_type(4)));

typedef __attribute__((address_space(1))) int* gint_p;   // b32 DMA src
typedef __attribute__((address_space(3))) int* lint_p;   // b32 DMA dst
typedef __attribute__((address_space(1))) i4*  gi4_p;    // b128 DMA src
typedef __attribute__((address_space(3))) i4*  li4_p;    // b128 DMA dst

__device__ __forceinline__ f4 fmin4(f4 a, f4 b) {
  f4 r;
  r.x = fminf(a.x, b.x);
  r.y = fminf(a.y, b.y);
  r.z = fminf(a.z, b.z);
  r.w = fminf(a.w, b.w);
  return r;
}

__global__ __launch_bounds__(NTHREADS) void erosion5x5_kernel(
    const float* __restrict__ in, float* __restrict__ out) {
  // row stride 136 floats = 544 B (16B multiple -> aligned float4 rows)
  __shared__ float raw[LH * LW];    // 10880 B staged halo tile
  __shared__ float hbuf[LH * TW];   // 10240 B horizontal-min intermediate

  const int tid = threadIdx.x;
  const int tx0 = blockIdx.x * TW;
  const int ty0 = blockIdx.y * TH;
  const int img = blockIdx.z;

  const float* __restrict__ src = in + (size_t)img * (HW_ * HW_);
  float* __restrict__ dst = out + (size_t)img * (HW_ * HW_);

  // ---- Stage 0: async-DMA halo tile into LDS; OOB chunks -> MAXV ----------
  // Chunk c of row ly covers gx = tx0-4+4c .. +3; always fully in or out.
  for (int e = tid; e < LH * NCHUNK; e += NTHREADS) {
    const int c  = e % NCHUNK;
    const int ly = e / NCHUNK;
    const int lx0 = 4 * c;
    const int gx0 = tx0 - 4 + lx0;
    const int gy  = ty0 + ly - P;
    const bool inb = ((unsigned)gx0 < (unsigned)HW_) &&   // gx0 in [0,1020]
                     ((unsigned)gy < (unsigned)HW_);
    float* lds_p = &raw[ly * LW + lx0];
#if ASYNC_MODE == 2
    if (inb) {
      __builtin_amdgcn_global_load_async_to_lds_b128(
          (gi4_p)(src + (size_t)gy * HW_ + gx0), (li4_p)lds_p, 0, 0);
    } else {
      *reinterpret_cast<f4*>(lds_p) = (f4){MAXV, MAXV, MAXV, MAXV};
    }
#elif ASYNC_MODE == 1
    if (inb) {
      const float* g = src + (size_t)gy * HW_ + gx0;
#pragma unroll
      for (int k = 0; k < 4; ++k)
        __builtin_amdgcn_global_load_async_to_lds_b32((gint_p)(g + k),
                                                      (lint_p)(lds_p + k), 0, 0);
    } else {
      *reinterpret_cast<f4*>(lds_p) = (f4){MAXV, MAXV, MAXV, MAXV};
    }
#else
    *reinterpret_cast<f4*>(lds_p) =
        inb ? *reinterpret_cast<const f4*>(src + (size_t)gy * HW_ + gx0)
            : (f4){MAXV, MAXV, MAXV, MAXV};
#endif
  }

#if ASYNC_MODE > 0
#if __has_builtin(__builtin_amdgcn_s_wait_asynccnt)
  __builtin_amdgcn_s_wait_asynccnt(0);
#else
  asm volatile("s_wait_asynccnt 0" ::: "memory");
#endif
#endif
  __syncthreads();

  // ---- Stage 1: horizontal 5-min, 4 outputs/thread via 3x ds_load_b128 ----
  // Output col x (0..127) needs raw lx = x+2 .. x+6.
  for (int e = tid; e < LH * (TW / 4); e += NTHREADS) {  // 640 float4 tasks
    const int x4 = (e % (TW / 4)) * 4;
    const int y  = e / (TW / 4);
    const float* __restrict__ rb = &raw[y * LW + x4];
    const f4 a  = *reinterpret_cast<const f4*>(rb);      // lx = x4+0..3
    const f4 b  = *reinterpret_cast<const f4*>(rb + 4);  // lx = x4+4..7
    const f4 cc = *reinterpret_cast<const f4*>(rb + 8);  // lx = x4+8..11
    // r[i] = raw[y][x4+i], need windows r[2..6], r[3..7], r[4..8], r[5..9]
    f4 o;
    o.x = fminf(fminf(fminf(a.z, a.w), fminf(b.x, b.y)), b.z);
    o.y = fminf(fminf(fminf(a.w, b.x), fminf(b.y, b.z)), b.w);
    o.z = fminf(fminf(fminf(b.x, b.y), fminf(b.z, b.w)), cc.x);
    o.w = fminf(fminf(fminf(b.y, b.z), fminf(b.w, cc.x)), cc.y);
    *reinterpret_cast<f4*>(&hbuf[y * TW + x4]) = o;
  }
  __syncthreads();

  // ---- Stage 2: vertical 5-min on float4 + b128 NT streaming stores -------
  // 32 lanes x float4 = one 128-col row segment; 8 rows per pass, 2 passes.
  {
    const int x4 = (tid % (TW / 4)) * 4;
    const int yb = tid / (TW / 4);  // 0..7
#pragma unroll
    for (int it = 0; it < 2; ++it) {
      const int y = yb + it * 8;  // output row in tile, hbuf rows y..y+4
      const f4 r0 = *reinterpret_cast<const f4*>(&hbuf[(y + 0) * TW + x4]);
      const f4 r1 = *reinterpret_cast<const f4*>(&hbuf[(y + 1) * TW + x4]);
      const f4 r2 = *reinterpret_cast<const f4*>(&hbuf[(y + 2) * TW + x4]);
      const f4 r3 = *reinterpret_cast<const f4*>(&hbuf[(y + 3) * TW + x4]);
      const f4 r4 = *reinterpret_cast<const f4*>(&hbuf[(y + 4) * TW + x4]);
      const f4 m = fmin4(fmin4(fmin4(r0, r1), fmin4(r2, r3)), r4);
      // write-once streaming output; bypass caches (working set > 192MB L2)
      __builtin_nontemporal_store(
          m, reinterpret_cast<f4*>(&dst[(size_t)(ty0 + y) * HW_ + tx0 + x4]));
    }
  }
}

extern "C" void kernel_launch(void* const* d_in, const int* in_sizes, int n_in,
                              void* d_out, int out_size, void* d_ws,
                              size_t ws_size, hipStream_t stream) {
  (void)n_in;
  (void)d_ws;
  (void)ws_size;
  (void)out_size;

  const float* in = (const float*)d_in[0];
  float* out = (float*)d_out;

  const int nc = in_sizes[0] / (HW_ * HW_);  // 32 images
  dim3 grid(HW_ / TW, HW_ / TH, nc);         // 8 x 64 x 32 blocks
  erosion5x5_kernel<<<grid, NTHREADS, 0, stream>>>(in, out);
}